// ObjectTokenExtractor_17446157156783
// MI455X (gfx1250) — compile-verified
//
#include <hip/hip_runtime.h>
#include <cstddef>

// ---- problem constants (DINOv2 ViT-L/14 @ 518) ----
#define B_    64
#define T_    10
#define D_    1024
#define GH_   37
#define GW_   37
#define G_    (GH_ * GW_)          // 1369
#define GPAD_ 1372                 // G rounded up to multiple of 4
#define KMAIN_ 1368                // last full-in-range k step starts below this
#define W_    518.0f
#define H_    518.0f
#define EFFW_ 518.0f
#define PW_   14.0f                // 518/37
#define PH_   14.0f

// ---- flat output offsets (floats), tuple return order ----
#define O_CLS    0
#define O_OBJ    (O_CLS + B_ * D_)                 //   65536
#define O_MASK   (O_OBJ + B_ * T_ * D_)            //  720896
#define O_BOXES  (O_MASK + B_ * T_)                //  721536
#define O_SCORES (O_BOXES + B_ * T_ * 4)           //  724096
#define O_ATTN   (O_SCORES + B_ * T_)              //  724736

typedef __attribute__((ext_vector_type(2))) float v2f;
typedef __attribute__((ext_vector_type(8))) float v8f;

// ---------------------------------------------------------------------------
// Kernel 1: logits[b,g] = dot(patch[b,g,:], attn_w) + attn_b
// One wave32 per row; float4 coalesced loads (D=1024 -> 8 x float4 per lane).
// ---------------------------------------------------------------------------
__global__ void logits_kernel(const float* __restrict__ patch,
                              const float* __restrict__ attn_w,
                              const float* __restrict__ attn_b,
                              float* __restrict__ logits) {
    const int lane = threadIdx.x & 31;
    const int wave = threadIdx.x >> 5;
    const long long row = (long long)blockIdx.x * 8 + wave;   // b*G + g
    if (row >= (long long)B_ * G_) return;

    const float4* p = (const float4*)(patch + row * D_);
    const float4* w = (const float4*)attn_w;
    float acc = 0.0f;
#pragma unroll
    for (int i = 0; i < 8; ++i) {
        const float4 pv = p[i * 32 + lane];
        const float4 wv = w[i * 32 + lane];
        acc = fmaf(pv.x, wv.x, acc);
        acc = fmaf(pv.y, wv.y, acc);
        acc = fmaf(pv.z, wv.z, acc);
        acc = fmaf(pv.w, wv.w, acc);
    }
#pragma unroll
    for (int off = 16; off > 0; off >>= 1) acc += __shfl_xor(acc, off, 32);
    if (lane == 0) logits[row] = acc + attn_b[0];
}

// ---------------------------------------------------------------------------
// Kernel 2: per-(b,t) rectangular masked softmax -> attention_maps, plus the
// small padded outputs. One wave32 per (b,t); wave-local shfl reductions only.
// ---------------------------------------------------------------------------
__global__ void softmax_kernel(const float* __restrict__ logits,
                               const float* __restrict__ boxes,
                               const float* __restrict__ scores,
                               const int* __restrict__ num_boxes,
                               float* __restrict__ out) {
    const int lane = threadIdx.x & 31;
    const int wave = threadIdx.x >> 5;
    const int bt = blockIdx.x * 8 + wave;
    if (bt >= B_ * T_) return;
    const int b = bt / T_;
    const int t = bt - b * T_;
    const int nb = num_boxes[b];
    const bool valid = (t < nb);
    const bool no_obj = (nb == 0);

    // small outputs
    if (lane < 4) out[O_BOXES + bt * 4 + lane] = valid ? boxes[bt * 4 + lane] : 0.0f;
    if (lane == 0) {
        float m = valid ? 1.0f : 0.0f;
        float s = valid ? scores[bt] : 0.0f;
        if (t == 0 && no_obj) { m = 1.0f; s = 1.0f; }
        out[O_MASK + bt] = m;
        out[O_SCORES + bt] = s;
    }

    float* attn = out + O_ATTN + (size_t)bt * G_;
    if (!valid) {
        for (int g = lane; g < G_; g += 32) attn[g] = 0.0f;
        return;
    }

    // box -> patch-index rectangle (same clamping as reference)
    const float bx0 = boxes[bt * 4 + 0], by0 = boxes[bt * 4 + 1];
    const float bx1 = boxes[bt * 4 + 2], by1 = boxes[bt * 4 + 3];
    const float x0 = fminf(fmaxf(bx0 * W_, 0.0f), EFFW_);
    const float y0 = fminf(fmaxf(by0 * H_, 0.0f), H_);
    const float x1 = fminf(fmaxf(bx1 * W_, 0.0f), EFFW_);
    const float y1 = fminf(fmaxf(by1 * H_, 0.0f), H_);
    const int x0i = min(max((int)floorf(x0 / PW_), 0), GW_ - 1);
    const int y0i = min(max((int)floorf(y0 / PH_), 0), GH_ - 1);
    const int x1i = min(max((int)ceilf(x1 / PW_), x0i + 1), GW_);
    const int y1i = min(max((int)ceilf(y1 / PH_), y0i + 1), GH_);

    const float* lg = logits + (size_t)b * G_;

    // pass 1: max over in-rect logits (rect always holds >= 1 patch)
    float mx = -3.402823466e+38f;
    for (int g = lane; g < G_; g += 32) {
        const int gy = g / GW_, gx = g - gy * GW_;
        if (gy >= y0i && gy < y1i && gx >= x0i && gx < x1i) mx = fmaxf(mx, lg[g]);
    }
#pragma unroll
    for (int off = 16; off > 0; off >>= 1) mx = fmaxf(mx, __shfl_xor(mx, off, 32));

    // pass 2: sum of exp
    float sm = 0.0f;
    for (int g = lane; g < G_; g += 32) {
        const int gy = g / GW_, gx = g - gy * GW_;
        if (gy >= y0i && gy < y1i && gx >= x0i && gx < x1i) sm += expf(lg[g] - mx);
    }
#pragma unroll
    for (int off = 16; off > 0; off >>= 1) sm += __shfl_xor(sm, off, 32);
    const float inv = 1.0f / sm;

    // pass 3: write weights (zeros outside rect)
    for (int g = lane; g < G_; g += 32) {
        const int gy = g / GW_, gx = g - gy * GW_;
        const bool in = (gy >= y0i && gy < y1i && gx >= x0i && gx < x1i);
        attn[g] = in ? expf(lg[g] - mx) * inv : 0.0f;
    }
}

// ---------------------------------------------------------------------------
// Kernel 3: pooled[b] = weights[b] (T x G) @ patch[b] (G x D) via
// V_WMMA_F32_16X16X4_F32 (full-f32 matrix core, exact math).
//   grid = (B, 4), block = 256 (8 waves). Block covers 256 of D's columns;
//   each wave owns two 16-wide N tiles.
// Hot loop is branch-free:
//   - A rows >= T: row index clamped to 0, value multiplied by a 0/1 mask
//     (2 v_mul instead of predicated ds_loads).
//   - K tail (g >= G): handled by one epilogue step whose out-of-range B rows
//     are address-clamped to row G-1; contribution killed by zero-padded A
//     columns (weights are finite, 0 * finite == 0).
// VGPR layouts per ISA 7.12.2 (m = lane&15, hi = lane>>4):
//   A v2f: { A[m][k+2*hi], A[m][k+2*hi+1] }   (one ds_load_b64, 8B aligned)
//   B v2f: { B[k+2*hi][n], B[k+2*hi+1][n] }
//   C v8f: c[r] -> (M = r + 8*hi, N = n0 + m)
// ---------------------------------------------------------------------------
__global__ void pool_wmma_kernel(const float* __restrict__ patch,
                                 const float* __restrict__ attn,   // [B,T,G]
                                 float* __restrict__ obj) {        // [B,T,D]
    __shared__ float wlds[T_][GPAD_];            // 10 * 1372 * 4 = 54880 B
    const int b = blockIdx.x;
    const float* am = attn + (size_t)b * T_ * G_;

    // stage weights; zero-pad columns G_..GPAD_-1
    for (int i = threadIdx.x; i < T_ * GPAD_; i += blockDim.x) {
        const int t = i / GPAD_, g = i - t * GPAD_;
        wlds[t][g] = (g < G_) ? am[t * G_ + g] : 0.0f;
    }
    __syncthreads();

    const int lane = threadIdx.x & 31;
    const int wave = threadIdx.x >> 5;
    const int m    = lane & 15;
    const int hi   = lane >> 4;
    const int row  = (m < T_) ? m : 0;                 // clamped A row
    const float amask = (m < T_) ? 1.0f : 0.0f;        // kills rows 10..15
    const int col0 = blockIdx.y * 256 + wave * 32;     // first N tile base
    const float* Pb = patch + (size_t)b * G_ * D_;

    // per-lane pointers (pure induction in the loop)
    const float* ap = &wlds[row][2 * hi];                          // LDS, 8B aligned
    const float* pb = Pb + (size_t)(2 * hi) * D_ + col0 + m;       // global

    v8f c0 = {};
    v8f c1 = {};
    for (int k = 0; k < KMAIN_; k += 4) {              // 342 branch-free iters
        const v2f aw = *(const v2f*)ap;
        v2f av; av.x = aw.x * amask; av.y = aw.y * amask;
        v2f bv0, bv1;
        bv0.x = pb[0];        bv0.y = pb[D_];
        bv1.x = pb[16];       bv1.y = pb[D_ + 16];
        c0 = __builtin_amdgcn_wmma_f32_16x16x4_f32(false, av, false, bv0,
                                                   (short)0, c0, false, false);
        c1 = __builtin_amdgcn_wmma_f32_16x16x4_f32(false, av, false, bv1,
                                                   (short)0, c1, false, false);
        ap += 4;
        pb += 4 * (size_t)D_;
    }

    // K tail: k = 1368. Valid g: 1368 only. A cols 1369..1371 are zero in LDS,
    // so clamp all B rows to 1368; bogus products are exactly zero.
    {
        const v2f aw = *(const v2f*)&wlds[row][KMAIN_ + 2 * hi];
        v2f av; av.x = aw.x * amask; av.y = aw.y * amask;
        const float* pt = Pb + (size_t)(G_ - 1) * D_ + col0 + m;   // row 1368
        v2f bv0; bv0.x = pt[0];  bv0.y = pt[0];
        v2f bv1; bv1.x = pt[16]; bv1.y = pt[16];
        c0 = __builtin_amdgcn_wmma_f32_16x16x4_f32(false, av, false, bv0,
                                                   (short)0, c0, false, false);
        c1 = __builtin_amdgcn_wmma_f32_16x16x4_f32(false, av, false, bv1,
                                                   (short)0, c1, false, false);
    }

#pragma unroll
    for (int r = 0; r < 8; ++r) {
        const int t = r + 8 * hi;
        if (t < T_) {
            obj[((size_t)b * T_ + t) * D_ + col0 + m]      = c0[r];
            obj[((size_t)b * T_ + t) * D_ + col0 + 16 + m] = c1[r];
        }
    }
}

// ---------------------------------------------------------------------------
// Kernel 4: cls passthrough + cls fallback into object_tokens[:,0,:] when an
// image has no boxes. Runs after pooling (stream order).
// ---------------------------------------------------------------------------
__global__ void fixup_kernel(const float* __restrict__ cls,
                             const int* __restrict__ num_boxes,
                             float* __restrict__ out) {
    const int i = blockIdx.x * blockDim.x + threadIdx.x;   // over B*D
    if (i >= B_ * D_) return;
    const int b = i / D_;
    const int d = i - b * D_;
    const float v = cls[i];
    out[O_CLS + i] = v;
    if (num_boxes[b] == 0) out[O_OBJ + ((size_t)b * T_) * D_ + d] = v;
}

// ---------------------------------------------------------------------------
extern "C" void kernel_launch(void* const* d_in, const int* in_sizes, int n_in,
                              void* d_out, int out_size, void* d_ws, size_t ws_size,
                              hipStream_t stream) {
    const float* cls       = (const float*)d_in[0];
    const float* patch     = (const float*)d_in[1];
    const float* boxes     = (const float*)d_in[2];
    const float* scores    = (const float*)d_in[3];
    const float* attn_w    = (const float*)d_in[4];
    const float* attn_b    = (const float*)d_in[5];
    const int*   num_boxes = (const int*)d_in[6];
    float* out    = (float*)d_out;
    float* logits = (float*)d_ws;                 // B*G floats = 350 KB

    (void)in_sizes; (void)n_in; (void)out_size; (void)ws_size;

    const int rows = B_ * G_;                     // 87616 logit rows
    logits_kernel<<<(rows + 7) / 8, 256, 0, stream>>>(patch, attn_w, attn_b, logits);

    softmax_kernel<<<(B_ * T_ + 7) / 8, 256, 0, stream>>>(logits, boxes, scores,
                                                          num_boxes, out);

    pool_wmma_kernel<<<dim3(B_, 4), 256, 0, stream>>>(patch, out + O_ATTN, out + O_OBJ);

    fixup_kernel<<<(B_ * D_ + 255) / 256, 256, 0, stream>>>(cls, num_boxes, out);
}